// DecoderRNN_3813930959215
// MI455X (gfx1250) — compile-verified
//
#include <hip/hip_runtime.h>
#include <math.h>

// Problem sizes (fixed by the reference)
#define BB 64
#define EE 2048
#define HH 1024
#define SS 512
#define VV 32000

typedef float v2f __attribute__((ext_vector_type(2)));
typedef float v8f __attribute__((ext_vector_type(8)));

// ---------------------------------------------------------------------------
// fp32 WMMA helper: D = A(16x4) * B(4x16) + C, exact fp32 semantics.
// ---------------------------------------------------------------------------
__device__ __forceinline__ v8f wmma4(v2f a, v2f b, v8f c) {
    return __builtin_amdgcn_wmma_f32_16x16x4_f32(
        false, a, false, b, (short)0, c, false, false);
}

// A fragment: A row-major [*, lda]. lane = (half<<4)|idx holds
// A[mBase+idx][kBase+2*half .. +1] -> contiguous 8B load. A is L2-resident
// and heavily reused -> regular-temporal loads.
__device__ __forceinline__ v2f ldA(const float* __restrict__ A, int lda,
                                   int mBase, int kBase, int idx, int half) {
    const float* p = A + (size_t)(mBase + idx) * lda + (kBase + 2 * half);
    return *(const v2f*)p;
}

// B fragment: weight matrices are streamed exactly once -> non-temporal
// loads (TH=NT) so the 262MB Wout / 63MB gate-weight streams don't evict
// the reusable working set (sv_emb, acat, x, m) from the 192MB L2.
__device__ __forceinline__ v2f ldB_nt(const float* __restrict__ Bm, int ldb,
                                      int kBase, int nBase, int idx, int half) {
    const float* p = Bm + (size_t)(kBase + 2 * half) * ldb + (nBase + idx);
    v2f r;
    r.x = __builtin_nontemporal_load(p);
    r.y = __builtin_nontemporal_load(p + ldb);
    return r;
}

// ---------------------------------------------------------------------------
// K1: m = (x@Wmx + bmx) * (h0@Wmh + bmh)            m:[64,1024]
// grid = H/16 N-tiles, 4 waves/block = 4 M-tiles (M=64)
// ---------------------------------------------------------------------------
__global__ void k_gemm_m(const float* __restrict__ x, const float* __restrict__ h0,
                         const float* __restrict__ Wmx, const float* __restrict__ bmx,
                         const float* __restrict__ Wmh, const float* __restrict__ bmh,
                         float* __restrict__ m_out) {
    const int lane = threadIdx.x & 31;
    const int wave = threadIdx.x >> 5;
    const int idx = lane & 15, half = lane >> 4;
    const int nBase = blockIdx.x * 16;
    const int mBase = wave * 16;

    v8f accx = {};
#pragma unroll 4
    for (int k = 0; k < EE; k += 4)
        accx = wmma4(ldA(x, EE, mBase, k, idx, half),
                     ldB_nt(Wmx, HH, k, nBase, idx, half), accx);

    v8f acch = {};
#pragma unroll 4
    for (int k = 0; k < HH; k += 4)
        acch = wmma4(ldA(h0, HH, mBase, k, idx, half),
                     ldB_nt(Wmh, HH, k, nBase, idx, half), acch);

    const float bx_ = bmx[nBase + idx];
    const float bh_ = bmh[nBase + idx];
#pragma unroll
    for (int r = 0; r < 8; ++r) {
        const int row = mBase + r + 8 * half;
        m_out[(size_t)row * HH + nBase + idx] = (accx[r] + bx_) * (acch[r] + bh_);
    }
}

// ---------------------------------------------------------------------------
// K2: g = x@Wx + m@Wm + bx + bm                      g:[64,4096]
// grid = 4H/16 = 256 N-tiles, 4 waves/block = M tiles
// ---------------------------------------------------------------------------
__global__ void k_gemm_g(const float* __restrict__ x, const float* __restrict__ m,
                         const float* __restrict__ Wx, const float* __restrict__ bx,
                         const float* __restrict__ Wm, const float* __restrict__ bm,
                         float* __restrict__ g) {
    const int lane = threadIdx.x & 31;
    const int wave = threadIdx.x >> 5;
    const int idx = lane & 15, half = lane >> 4;
    const int nBase = blockIdx.x * 16;
    const int mBase = wave * 16;
    const int ldg = 4 * HH;

    v8f acc = {};
#pragma unroll 4
    for (int k = 0; k < EE; k += 4)
        acc = wmma4(ldA(x, EE, mBase, k, idx, half),
                    ldB_nt(Wx, ldg, k, nBase, idx, half), acc);
#pragma unroll 4
    for (int k = 0; k < HH; k += 4)
        acc = wmma4(ldA(m, HH, mBase, k, idx, half),
                    ldB_nt(Wm, ldg, k, nBase, idx, half), acc);

    const float bias = bx[nBase + idx] + bm[nBase + idx];
#pragma unroll
    for (int r = 0; r < 8; ++r) {
        const int row = mBase + r + 8 * half;
        g[(size_t)row * ldg + nBase + idx] = acc[r] + bias;
    }
}

// ---------------------------------------------------------------------------
// K3: gates -> h  (h written into first half of Acat[64,2048])
// ---------------------------------------------------------------------------
__global__ void k_gates(const float* __restrict__ g, const float* __restrict__ c0,
                        float* __restrict__ acat) {
    const int t = blockIdx.x * blockDim.x + threadIdx.x;   // 0 .. 64*1024-1
    const int b = t >> 10;
    const int j = t & 1023;
    const float* gr = g + (size_t)b * 4 * HH;
    const float f  = 1.0f / (1.0f + __expf(-gr[j]));
    const float i  = 1.0f / (1.0f + __expf(-gr[j + HH]));
    const float o  = 1.0f / (1.0f + __expf(-gr[j + 2 * HH]));
    const float ct = tanhf(gr[j + 3 * HH]);
    const float c  = f * c0[t] + i * ct;
    acat[(size_t)b * (2 * HH) + j] = o * tanhf(c);
}

// ---------------------------------------------------------------------------
// K4: attention. One block per batch row. h cached in LDS, scores+softmax
// in LDS, context = attn @ sv_emb with coalesced column-streaming.
// sv_emb (134MB) is read twice and fits in the 192MB L2, so pass 2 should
// hit L2 -> keep these loads regular-temporal.
// ---------------------------------------------------------------------------
__global__ void k_attn(float* __restrict__ acat, const float* __restrict__ sv) {
    __shared__ float sc[SS];
    __shared__ float hsh[HH];
    __shared__ float red[1];
    const int b = blockIdx.x;
    const float* svb = sv + (size_t)b * SS * HH;
    const float* hb  = acat + (size_t)b * (2 * HH);

    for (int d = threadIdx.x; d < HH; d += blockDim.x) hsh[d] = hb[d];
    __syncthreads();

    const int lane = threadIdx.x & 31;
    const int wave = threadIdx.x >> 5;            // 8 waves
    for (int s = wave; s < SS; s += 8) {          // wave-uniform s
        const float* row = svb + (size_t)s * HH;
        float p = 0.0f;
        for (int d = lane; d < HH; d += 32) p += hsh[d] * row[d];
        for (int off = 16; off; off >>= 1) p += __shfl_xor(p, off, 32);
        if (lane == 0) sc[s] = p;
    }
    __syncthreads();

    if (threadIdx.x < 32) {                       // softmax stats in one wave
        float mx = -3.0e38f;
        for (int s = lane; s < SS; s += 32) mx = fmaxf(mx, sc[s]);
        for (int off = 16; off; off >>= 1) mx = fmaxf(mx, __shfl_xor(mx, off, 32));
        float sum = 0.0f;
        for (int s = lane; s < SS; s += 32) {
            const float e = __expf(sc[s] - mx);
            sc[s] = e;
            sum += e;
        }
        for (int off = 16; off; off >>= 1) sum += __shfl_xor(sum, off, 32);
        if (lane == 0) red[0] = 1.0f / sum;
    }
    __syncthreads();

    const float inv = red[0];
    float* ctx = acat + (size_t)b * (2 * HH) + HH;
    for (int d = threadIdx.x; d < HH; d += blockDim.x) {
        float a = 0.0f;
        for (int s = 0; s < SS; ++s) a += sc[s] * svb[(size_t)s * HH + d];
        ctx[d] = a * inv;
    }
}

// ---------------------------------------------------------------------------
// K5: logits = Acat[64,2048] @ Wout[2048,32000] + bout  (the 262MB stream)
// 8 waves/block, wave owns one N-tile and ALL 4 M-tiles so each Wout
// fragment is loaded exactly once and feeds 4 WMMAs. Acat comes from L2.
// Wout loads NT (single use, > L2 size); logits stores NT (write-once).
// ---------------------------------------------------------------------------
__global__ void k_logits(const float* __restrict__ acat, const float* __restrict__ Wout,
                         const float* __restrict__ bout, float* __restrict__ out) {
    const int lane = threadIdx.x & 31;
    const int wave = threadIdx.x >> 5;
    const int idx = lane & 15, half = lane >> 4;
    const int nBase = (blockIdx.x * 8 + wave) * 16;
    const int K = 2 * HH;

    v8f a0 = {}, a1 = {}, a2 = {}, a3 = {};
#pragma unroll 2
    for (int k = 0; k < K; k += 4) {
        if (k + 64 < K)   // uniform guard; pull the Wout stream ahead
            __builtin_prefetch(Wout + (size_t)(k + 64 + 2 * half) * VV + nBase + idx, 0, 1);
        const v2f bf = ldB_nt(Wout, VV, k, nBase, idx, half);
        a0 = wmma4(ldA(acat, K, 0, k, idx, half), bf, a0);
        a1 = wmma4(ldA(acat, K, 16, k, idx, half), bf, a1);
        a2 = wmma4(ldA(acat, K, 32, k, idx, half), bf, a2);
        a3 = wmma4(ldA(acat, K, 48, k, idx, half), bf, a3);
    }

    const float bb = bout[nBase + idx];
#pragma unroll
    for (int r = 0; r < 8; ++r) {
        const int row = r + 8 * half;
        __builtin_nontemporal_store(a0[r] + bb, out + (size_t)(row) * VV + nBase + idx);
        __builtin_nontemporal_store(a1[r] + bb, out + (size_t)(row + 16) * VV + nBase + idx);
        __builtin_nontemporal_store(a2[r] + bb, out + (size_t)(row + 32) * VV + nBase + idx);
        __builtin_nontemporal_store(a3[r] + bb, out + (size_t)(row + 48) * VV + nBase + idx);
    }
}

// ---------------------------------------------------------------------------
extern "C" void kernel_launch(void* const* d_in, const int* in_sizes, int n_in,
                              void* d_out, int out_size, void* d_ws, size_t ws_size,
                              hipStream_t stream) {
    const float* x    = (const float*)d_in[0];
    const float* h0   = (const float*)d_in[1];
    const float* c0   = (const float*)d_in[2];
    const float* sv   = (const float*)d_in[3];
    const float* Wmx  = (const float*)d_in[4];
    const float* bmx  = (const float*)d_in[5];
    const float* Wmh  = (const float*)d_in[6];
    const float* bmh  = (const float*)d_in[7];
    const float* Wx   = (const float*)d_in[8];
    const float* bx   = (const float*)d_in[9];
    const float* Wm   = (const float*)d_in[10];
    const float* bm   = (const float*)d_in[11];
    const float* Wout = (const float*)d_in[12];
    const float* bout = (const float*)d_in[13];
    float* out = (float*)d_out;

    float* ws    = (float*)d_ws;
    float* m_buf = ws;                              // 64*1024
    float* g_buf = ws + (size_t)BB * HH;            // 64*4096
    float* acat  = g_buf + (size_t)BB * 4 * HH;     // 64*2048  [h | context]

    k_gemm_m<<<dim3(HH / 16), 128, 0, stream>>>(x, h0, Wmx, bmx, Wmh, bmh, m_buf);
    k_gemm_g<<<dim3(4 * HH / 16), 128, 0, stream>>>(x, m_buf, Wx, bx, Wm, bm, g_buf);
    k_gates<<<dim3((BB * HH) / 256), 256, 0, stream>>>(g_buf, c0, acat);
    k_attn<<<dim3(BB), 256, 0, stream>>>(acat, sv);
    k_logits<<<dim3(VV / 16 / 8), 256, 0, stream>>>(acat, Wout, bout, out);
}